// AddSelfEnergies_18030272708652
// MI455X (gfx1250) — compile-verified
//
#include <hip/hip_runtime.h>

// CDNA5 / gfx1250: wave32, WMMA 16x16x4 f32 used as a deterministic
// 4-way-per-molecule cross-lane reduction (B = ones => D[m][*] = row-sum of A).

typedef __attribute__((ext_vector_type(2))) float v2f;
typedef __attribute__((ext_vector_type(8))) float v8f;

#define MOLS_PER_WAVE   16
#define WAVES_PER_BLOCK 8
#define BLOCK_THREADS   (WAVES_PER_BLOCK * 32)

__global__ __launch_bounds__(BLOCK_THREADS)
void self_energy_segsum_kernel(const float* __restrict__ energy_readout,
                               const int*   __restrict__ atomic_numbers,
                               const int*   __restrict__ subsys_idx,   // sorted
                               const float* __restrict__ se_tab,
                               float*       __restrict__ out,
                               int num_molecules, int num_atoms, int se_len)
{
    __shared__ float seTab[128];                       // padded self-energy LUT
    __shared__ float red[WAVES_PER_BLOCK][MOLS_PER_WAVE];

    // Stage the tiny LUT into LDS once per block.
    for (int t = threadIdx.x; t < 128; t += BLOCK_THREADS)
        seTab[t] = (t < se_len) ? se_tab[t] : 0.0f;
    __syncthreads();

    const int lane  = threadIdx.x & 31;
    const int wave  = threadIdx.x >> 5;
    const int gwave = blockIdx.x * WAVES_PER_BLOCK + wave;
    const int base  = gwave * MOLS_PER_WAVE;
    if (base >= num_molecules) return;                 // wave-uniform

    // --- segment boundaries via binary search (indices are sorted) ---
    // lane L (L<=16) computes lower_bound(base + L); lanes 17..31 mirror lane 16.
    int target = base + (lane < 16 ? lane : 16);
    if (target > num_molecules) target = num_molecules;
    int lo = 0, hi = num_atoms;
    while (lo < hi) {
        int mid = (lo + hi) >> 1;
        if (subsys_idx[mid] < target) lo = mid + 1; else hi = mid;
    }
    const int j        = lane & 15;                    // molecule slot (M row)
    const int segStart = __shfl(lo, j,     32);
    const int segEnd   = __shfl(lo, j + 1, 32);        // j=15 -> lane16 = lb(base+16)
    const int hiHalf   = lane >> 4;                    // lanes L / L+16 split even/odd atoms

    // --- accumulate 2 partials per lane => 4 f32 partials per molecule ---
    float acc0 = 0.0f, acc1 = 0.0f;
    int i = segStart + hiHalf;
    for (; i + 2 < segEnd; i += 4) {
        acc0 += seTab[atomic_numbers[i]     & 127];
        acc1 += seTab[atomic_numbers[i + 2] & 127];
    }
    if (i < segEnd)
        acc0 += seTab[atomic_numbers[i] & 127];

    // --- WMMA reduction: A(16x4 f32) holds the partials per ISA layout:
    //     lanes 0-15 -> (M=lane, K=0/1), lanes 16-31 -> (M=lane-16, K=2/3).
    //     B = ones(4x16) => D[m][n] = sum_k A[m][k] = molecule sum, all lanes active.
    v2f A = {acc0, acc1};
    v2f B = {1.0f, 1.0f};
    v8f C = {};
    v8f D = __builtin_amdgcn_wmma_f32_16x16x4_f32(
        /*neg_a=*/false, A, /*neg_b=*/false, B,
        /*c_mod=*/(short)0, C, /*reuse_a=*/false, /*reuse_b=*/false);

    // D layout: lane0 VGPR r = row r (mol base+r), lane16 VGPR r = row 8+r.
    if ((lane & 15) == 0) {
        const int off = hiHalf * 8;
#pragma unroll
        for (int r = 0; r < 8; ++r)
            red[wave][off + r] = D[r];
    }
    __builtin_amdgcn_wave_barrier();                   // keep LDS store->load ordered

    if (lane < 16) {
        const int m = base + lane;
        if (m < num_molecules)
            out[m] = energy_readout[m] + red[wave][lane];
    }
}

extern "C" void kernel_launch(void* const* d_in, const int* in_sizes, int n_in,
                              void* d_out, int out_size, void* d_ws, size_t ws_size,
                              hipStream_t stream)
{
    (void)n_in; (void)out_size; (void)d_ws; (void)ws_size;
    const float* energy_readout = (const float*)d_in[0];
    const int*   atomic_numbers = (const int*)  d_in[1];
    const int*   subsys_idx     = (const int*)  d_in[2];
    const float* se_tab         = (const float*)d_in[3];
    float*       out            = (float*)d_out;

    const int num_molecules = in_sizes[0];
    const int num_atoms     = in_sizes[1];
    const int se_len        = in_sizes[3];

    const int waves  = (num_molecules + MOLS_PER_WAVE - 1) / MOLS_PER_WAVE;
    const int blocks = (waves + WAVES_PER_BLOCK - 1) / WAVES_PER_BLOCK;

    self_energy_segsum_kernel<<<blocks, BLOCK_THREADS, 0, stream>>>(
        energy_readout, atomic_numbers, subsys_idx, se_tab, out,
        num_molecules, num_atoms, se_len);
}